// BipartiteGraphConv_13778255086103
// MI455X (gfx1250) — compile-verified
//
#include <hip/hip_runtime.h>
#include <hip/hip_bf16.h>

// ---------------------------------------------------------------------------
// Bipartite graph conv, restructured: scatter-first, GEMM-second.
//   A_u = segsum(norm * v_e, edge_u)   B_u = segsum(norm * u_e*v_e, edge_u)
//   A_v = segsum(norm * u_e, edge_v)   B_v = segsum(norm * u_e*v_e, edge_v)
//   U_new = leaky((U + A_u) @ W1^T + B_u @ W2^T)   (same for V)
// Edge phase: HBM/atomic bound (~2 GB). Node GEMM: v_wmma_f32_16x16x4_f32
// (exact fp32 math, 9.8 GFLOP total -> negligible vs edge phase).
// ---------------------------------------------------------------------------

#define GDIM 128
#define LEAKY 0.1f
#define EPSV  1e-8f

typedef float v2f __attribute__((ext_vector_type(2)));
typedef float v8f __attribute__((ext_vector_type(8)));

// --------------------------- edge scatter kernel ---------------------------
// One wave32 per edge; each lane owns 4 contiguous channels (128/32 = 4).
__global__ __launch_bounds__(256) void edge_scatter_kernel(
    const float* __restrict__ U, const float* __restrict__ V,
    const long long* __restrict__ edge_u, const long long* __restrict__ edge_v,
    const float* __restrict__ deg_u, const float* __restrict__ deg_v,
    float* __restrict__ Au, float* __restrict__ Bu,
    float* __restrict__ Av, float* __restrict__ Bv, int nE)
{
    const int wave = threadIdx.x >> 5;          // 8 waves per block
    const int lane = threadIdx.x & 31;
    const int e = blockIdx.x * 8 + wave;        // wave-uniform edge id
    if (e >= nE) return;

    const long long eu = edge_u[e];
    const long long ev = edge_v[e];
    const float norm = rsqrtf(deg_u[eu] * deg_v[ev] + EPSV);

    const int c = lane * 4;
    const float4 u4 = *reinterpret_cast<const float4*>(U + (size_t)eu * GDIM + c);
    const float4 v4 = *reinterpret_cast<const float4*>(V + (size_t)ev * GDIM + c);

    float4 un, vn, uvn;
    un.x = u4.x * norm;  un.y = u4.y * norm;  un.z = u4.z * norm;  un.w = u4.w * norm;
    vn.x = v4.x * norm;  vn.y = v4.y * norm;  vn.z = v4.z * norm;  vn.w = v4.w * norm;
    uvn.x = un.x * v4.x; uvn.y = un.y * v4.y; uvn.z = un.z * v4.z; uvn.w = un.w * v4.w;

    float* au = Au + (size_t)eu * GDIM + c;
    float* bu = Bu + (size_t)eu * GDIM + c;
    float* av = Av + (size_t)ev * GDIM + c;
    float* bv = Bv + (size_t)ev * GDIM + c;

    unsafeAtomicAdd(au + 0, vn.x);  unsafeAtomicAdd(au + 1, vn.y);
    unsafeAtomicAdd(au + 2, vn.z);  unsafeAtomicAdd(au + 3, vn.w);
    unsafeAtomicAdd(bu + 0, uvn.x); unsafeAtomicAdd(bu + 1, uvn.y);
    unsafeAtomicAdd(bu + 2, uvn.z); unsafeAtomicAdd(bu + 3, uvn.w);
    unsafeAtomicAdd(av + 0, un.x);  unsafeAtomicAdd(av + 1, un.y);
    unsafeAtomicAdd(av + 2, un.z);  unsafeAtomicAdd(av + 3, un.w);
    unsafeAtomicAdd(bv + 0, uvn.x); unsafeAtomicAdd(bv + 1, uvn.y);
    unsafeAtomicAdd(bv + 2, uvn.z); unsafeAtomicAdd(bv + 3, uvn.w);
}

// ----------------------------- node GEMM kernel ----------------------------
// out[n, d] = leaky( sum_k (X+Aadd)[n,k]*W1[d,k] + Badd[n,k]*W2[d,k] )
// One wave computes a 16x128 output tile with V_WMMA_F32_16X16X4_F32.
// A-tile (16x4 f32, 2 VGPR): lanes 0-15 rows M=0..15 K={k,k+1};
//                            lanes 16-31 same rows K={k+2,k+3}.
// B-tile (4x16 f32, 2 VGPR): mirrored K-half split, N = lane&15.
// C/D (8 VGPR): element j -> M = j + 8*(lane>=16), N = lane&15.
__global__ __launch_bounds__(256) void node_gemm_kernel(
    const float* __restrict__ X, const float* __restrict__ Aadd,
    const float* __restrict__ Badd, const float* __restrict__ W1,
    const float* __restrict__ W2, float* __restrict__ out, int ntiles)
{
    const int wave = threadIdx.x >> 5;
    const int lane = threadIdx.x & 31;
    const int tile = blockIdx.x * 8 + wave;
    if (tile >= ntiles) return;

    const int row0  = tile * 16;
    const int ncol  = lane & 15;
    const int khalf = (lane >> 4) * 2;          // 0 for lanes 0-15, 2 for 16-31
    const int arow  = row0 + ncol;              // A-matrix row for this lane

    v8f acc[8];
    const v8f zero = {0.f, 0.f, 0.f, 0.f, 0.f, 0.f, 0.f, 0.f};
#pragma unroll
    for (int i = 0; i < 8; ++i) acc[i] = zero;

    const float* xrow = X    + (size_t)arow * GDIM + khalf;
    const float* arw  = Aadd + (size_t)arow * GDIM + khalf;
    const float* brw  = Badd + (size_t)arow * GDIM + khalf;

    for (int k = 0; k < GDIM; k += 4) {
        const float2 x2 = *reinterpret_cast<const float2*>(xrow + k);
        const float2 a2 = *reinterpret_cast<const float2*>(arw + k);
        const float2 b2 = *reinterpret_cast<const float2*>(brw + k);
        v2f amat1; amat1[0] = x2.x + a2.x; amat1[1] = x2.y + a2.y;  // (X+A) tile
        v2f amat2; amat2[0] = b2.x;        amat2[1] = b2.y;         // B tile

#pragma unroll
        for (int nt = 0; nt < 8; ++nt) {
            const int n = nt * 16 + ncol;       // output channel (W row)
            const float2 w1v = *reinterpret_cast<const float2*>(W1 + (size_t)n * GDIM + k + khalf);
            const float2 w2v = *reinterpret_cast<const float2*>(W2 + (size_t)n * GDIM + k + khalf);
            v2f bmat1; bmat1[0] = w1v.x; bmat1[1] = w1v.y;
            v2f bmat2; bmat2[0] = w2v.x; bmat2[1] = w2v.y;
            acc[nt] = __builtin_amdgcn_wmma_f32_16x16x4_f32(
                false, amat1, false, bmat1, (short)0, acc[nt], false, false);
            acc[nt] = __builtin_amdgcn_wmma_f32_16x16x4_f32(
                false, amat2, false, bmat2, (short)0, acc[nt], false, false);
        }
    }

    // Epilogue: leaky ReLU + store. VGPR j holds row M = j + 8*(lane>=16).
    const int mbase = (lane >> 4) * 8;
#pragma unroll
    for (int nt = 0; nt < 8; ++nt) {
#pragma unroll
        for (int j = 0; j < 8; ++j) {
            float x = acc[nt][j];
            x = (x >= 0.f) ? x : LEAKY * x;
            out[(size_t)(row0 + mbase + j) * GDIM + nt * 16 + ncol] = x;
        }
    }
}

// ------------------------------- launcher ----------------------------------
extern "C" void kernel_launch(void* const* d_in, const int* in_sizes, int n_in,
                              void* d_out, int out_size, void* d_ws, size_t ws_size,
                              hipStream_t stream) {
    const float*     U      = (const float*)d_in[0];
    const float*     V      = (const float*)d_in[1];
    const long long* edge_u = (const long long*)d_in[2];
    const long long* edge_v = (const long long*)d_in[3];
    const float*     deg_u  = (const float*)d_in[4];
    const float*     deg_v  = (const float*)d_in[5];
    const float*     W1     = (const float*)d_in[6];
    const float*     W2     = (const float*)d_in[7];

    const int nU = in_sizes[0] / GDIM;   // 100000
    const int nV = in_sizes[1] / GDIM;   // 50000
    const int nE = in_sizes[2];          // 640000

    // Workspace: four scatter accumulators (A_u, B_u, A_v, B_v) ~ 153.6 MB.
    float* Au = (float*)d_ws;
    float* Bu = Au + (size_t)nU * GDIM;
    float* Av = Bu + (size_t)nU * GDIM;
    float* Bv = Av + (size_t)nV * GDIM;
    const size_t accBytes = ((size_t)2 * nU + 2 * nV) * GDIM * sizeof(float);
    hipMemsetAsync(d_ws, 0, accBytes, stream);   // graph-capture-legal

    // Phase 1: edge gather/scale/scatter-add (one wave32 per edge).
    edge_scatter_kernel<<<(nE + 7) / 8, 256, 0, stream>>>(
        U, V, edge_u, edge_v, deg_u, deg_v, Au, Bu, Av, Bv, nE);

    // Phase 2: node-level dual-GEMM + leaky ReLU via f32 WMMA.
    float* Uout = (float*)d_out;
    float* Vout = Uout + (size_t)nU * GDIM;
    const int tilesU = nU / 16;   // 6250 (exact)
    const int tilesV = nV / 16;   // 3125 (exact)
    node_gemm_kernel<<<(tilesU + 7) / 8, 256, 0, stream>>>(U, Au, Bu, W1, W2, Uout, tilesU);
    node_gemm_kernel<<<(tilesV + 7) / 8, 256, 0, stream>>>(V, Av, Bv, W1, W2, Vout, tilesV);
}